// ts_corr_65455301591100
// MI455X (gfx1250) — compile-verified
//
#include <hip/hip_runtime.h>
#include <cstdint>

#define B_      128
#define T_      512
#define F_      40
#define W_      10
#define NS_     503          // number of window starts (stride 1; 502 included)
#define NPAIR_  780          // F*(F-1)/2
#define WCHUNK  64           // windows per block
#define NCHUNK  8            // ceil(503/64)
#define NWAVES  8            // 256 threads / wave32
#define EPS_    1e-5f

typedef float v2f __attribute__((ext_vector_type(2)));
typedef float v8f __attribute__((ext_vector_type(8)));

__global__ __launch_bounds__(256)
void ts_corr_kernel(const float* __restrict__ in, float* __restrict__ out)
{
    // slab of input rows covering this block's windows (contiguous in memory)
    __shared__ __attribute__((aligned(16))) float s_x[(WCHUNK + W_ - 1) * F_]; // 73*40
    __shared__ float s_mean[NWAVES * 48];
    __shared__ float s_rstd[NWAVES * 48];

    const int tid = threadIdx.x;
    const int b   = blockIdx.x / NCHUNK;
    const int c   = blockIdx.x % NCHUNK;
    const int s0  = c * WCHUNK;
    const int nwin  = min(WCHUNK, NS_ - s0);
    const int nrows = nwin + W_ - 1;                    // rows s0 .. s0+nrows-1

    // ---- async global->LDS staging copy (CDNA5 async-to-LDS path, 16B granules) ----
    {
        const int n4 = nrows * (F_ / 4);                // float4 granules (max 730)
        const uint64_t gbase = (uint64_t)(uintptr_t)(in + ((size_t)b * T_ + s0) * F_);
        const uint32_t lbase = (uint32_t)(uintptr_t)(&s_x[0]);  // LDS byte offset (addr[31:0])
        for (int i = tid; i < n4; i += 256) {
            const uint32_t laddr = lbase + (uint32_t)i * 16u;
            const uint64_t gaddr = gbase + (uint64_t)i * 16u;
            asm volatile("global_load_async_to_lds_b128 %0, %1, off"
                         :: "v"(laddr), "v"(gaddr) : "memory");
        }
        asm volatile("s_wait_asynccnt 0x0" ::: "memory");
    }
    __syncthreads();

    const int wave = tid >> 5;
    const int lane = tid & 31;
    const int half = lane >> 4;   // K-half for A/B fragment layout
    const int lm   = lane & 15;   // M/N index within tile

    float* meanW = s_mean + wave * 48;
    float* rstdW = s_rstd + wave * 48;

    // clamped per-tile feature index (always a valid LDS column)
    const int fidx0 = lm;
    const int fidx1 = 16 + lm;
    const int fidx2 = 32 + (lm < 8 ? lm : 7);
    const bool fok2 = (lm < 8);      // tile-2 feature in range (f < 40)
    const bool kok2 = (half == 0);   // ks==2: k = 8,9 valid; 10,11 padded

    for (int sw = wave; sw < nwin; sw += NWAVES) {
        const int s = s0 + sw;
        const float* xw = s_x + sw * F_;                // 10 rows x 40 feats

        // ---- per-feature mean / reciprocal-std into per-wave LDS scratch ----
        {
            const int f = lane;                         // 0..31, always < 40
            float sum = 0.f, ss = 0.f;
            #pragma unroll
            for (int w = 0; w < W_; ++w) { float v = xw[w * F_ + f]; sum += v; ss += v * v; }
            const float mu  = sum * 0.1f;
            const float var = ss * 0.1f - mu * mu;
            meanW[f] = mu;
            rstdW[f] = 1.0f / (sqrtf(fmaxf(var, 0.f)) + EPS_); // NaN->0 via fmax
            const int f2 = lane + 32;
            if (f2 < F_) {
                float sum2 = 0.f, ss2 = 0.f;
                #pragma unroll
                for (int w = 0; w < W_; ++w) { float v = xw[w * F_ + f2]; sum2 += v; ss2 += v * v; }
                const float mu2  = sum2 * 0.1f;
                const float var2 = ss2 * 0.1f - mu2 * mu2;
                meanW[f2] = mu2;
                rstdW[f2] = 1.0f / (sqrtf(fmaxf(var2, 0.f)) + EPS_);
            } else if (f2 < 48) {                       // pad features: finite dummies
                meanW[f2] = 0.f; rstdW[f2] = 1.f;
            }
        }

        // ---- fragments: frag[t][ks] element v holds X[k, f], k = ks*4+half*2+v,
        //      f = t*16+lm. Branchless: clamped address + v_cndmask zeroing. ----
        v2f frag[3][3];
        #pragma unroll
        for (int t = 0; t < 3; ++t) {
            const int fi = (t == 0) ? fidx0 : (t == 1) ? fidx1 : fidx2;
            const bool okf = (t < 2) || fok2;
            #pragma unroll
            for (int ks = 0; ks < 3; ++ks) {
                const bool okk = (ks < 2) || kok2;
                #pragma unroll
                for (int v = 0; v < 2; ++v) {
                    const int kc = (ks < 2) ? (ks * 4 + half * 2 + v)   // always < 10
                                            : (8 + v);                  // clamp padded half
                    const float val = xw[kc * F_ + fi];
                    frag[t][ks][v] = (okk && okf) ? val : 0.f;
                }
            }
        }

        // ---- 6 upper-triangle C tiles x 3 K-steps: 18 WMMAs / window ----
        v8f c00 = {}, c01 = {}, c02 = {}, c11 = {}, c12 = {}, c22 = {};
        #pragma unroll
        for (int ks = 0; ks < 3; ++ks) {
            c00 = __builtin_amdgcn_wmma_f32_16x16x4_f32(false, frag[0][ks], false, frag[0][ks], (short)0, c00, false, false);
            c01 = __builtin_amdgcn_wmma_f32_16x16x4_f32(false, frag[0][ks], false, frag[1][ks], (short)0, c01, false, false);
            c02 = __builtin_amdgcn_wmma_f32_16x16x4_f32(false, frag[0][ks], false, frag[2][ks], (short)0, c02, false, false);
            c11 = __builtin_amdgcn_wmma_f32_16x16x4_f32(false, frag[1][ks], false, frag[1][ks], (short)0, c11, false, false);
            c12 = __builtin_amdgcn_wmma_f32_16x16x4_f32(false, frag[1][ks], false, frag[2][ks], (short)0, c12, false, false);
            c22 = __builtin_amdgcn_wmma_f32_16x16x4_f32(false, frag[2][ks], false, frag[2][ks], (short)0, c22, false, false);
        }

        // ---- normalize + non-temporal scatter of packed upper triangle ----
        float* outw = out + ((size_t)b * NS_ + s) * NPAIR_;
        // per-column-tile terms (j depends only on lane & tj)
        float muj[3], rsj[3];
        #pragma unroll
        for (int t = 0; t < 3; ++t) { muj[t] = meanW[t * 16 + lm]; rsj[t] = rstdW[t * 16 + lm]; }

        #pragma unroll
        for (int ti = 0; ti < 3; ++ti) {
            float mui[8], rsi[8];
            #pragma unroll
            for (int v = 0; v < 8; ++v) {
                const int i = ti * 16 + half * 8 + v;
                mui[v] = meanW[i];
                rsi[v] = rstdW[i];
            }
            #pragma unroll
            for (int tj = ti; tj < 3; ++tj) {
                const v8f& acc = (ti == 0) ? (tj == 0 ? c00 : tj == 1 ? c01 : c02)
                               : (ti == 1) ? (tj == 1 ? c11 : c12)
                                           : c22;
                const int j = tj * 16 + lm;
                #pragma unroll
                for (int v = 0; v < 8; ++v) {
                    const int i = ti * 16 + half * 8 + v;
                    if (j > i && j < F_) {               // implies i < j < 40
                        const float corr = (acc[v] * 0.1f - mui[v] * muj[tj]) * (rsi[v] * rsj[tj]);
                        const int idx = i * F_ - (i * (i + 1)) / 2 + (j - i - 1);
                        __builtin_nontemporal_store(corr, &outw[idx]);
                    }
                }
            }
        }
    }
}

extern "C" void kernel_launch(void* const* d_in, const int* in_sizes, int n_in,
                              void* d_out, int out_size, void* d_ws, size_t ws_size,
                              hipStream_t stream)
{
    (void)in_sizes; (void)n_in; (void)out_size; (void)d_ws; (void)ws_size;
    const float* in = (const float*)d_in[0];
    float* out = (float*)d_out;
    ts_corr_kernel<<<dim3(B_ * NCHUNK), dim3(256), 0, stream>>>(in, out);
}